// EBSDCovmatKSphere_56925496541960
// MI455X (gfx1250) — compile-verified
//
#include <hip/hip_runtime.h>
#include <hip/hip_bf16.h>
#include <stdint.h>

typedef __attribute__((ext_vector_type(2))) float v2f;
typedef __attribute__((ext_vector_type(8))) float v8f;

#define N_QUERY 2048
#define M_FZ    4096
#define K_SYM   24
#define ROWS_PER_Q (2 * K_SYM)            // 48
#define NROWS   (N_QUERY * ROWS_PER_Q)    // 98304
#define NTILES  (NROWS / 16)              // 6144 row tiles
#define MTILES  (M_FZ / 16)               // 256 col tiles
#define WAVES_PER_BLOCK 8

// ---------------------------------------------------------------------------
// Kernel 1: quaternion-apply each query point by each Laue quat.
// Store per row: (-2x, -2y, -2z, a2)  so WMMA directly yields -2*dot and
// d2 = (a2 + b2) + acc matches the reference formula's association.
// ---------------------------------------------------------------------------
__global__ void prep_rows(const float* __restrict__ qpts,
                          const float* __restrict__ quats,
                          float4* __restrict__ eqiv) {
    int idx = blockIdx.x * blockDim.x + threadIdx.x;
    if (idx >= N_QUERY * K_SYM) return;
    int n = idx / K_SYM;
    int k = idx % K_SYM;

    float vx = qpts[n * 3 + 0], vy = qpts[n * 3 + 1], vz = qpts[n * 3 + 2];
    float w  = quats[k * 4 + 0];
    float qx = quats[k * 4 + 1], qy = quats[k * 4 + 2], qz = quats[k * 4 + 3];

    // uv = cross(q.xyz, v)
    float uvx = qy * vz - qz * vy;
    float uvy = qz * vx - qx * vz;
    float uvz = qx * vy - qy * vx;
    // uuv = cross(q.xyz, uv)
    float uuvx = qy * uvz - qz * uvy;
    float uuvy = qz * uvx - qx * uvz;
    float uuvz = qx * uvy - qy * uvx;

    float rx = vx + 2.0f * (w * uvx + uuvx);
    float ry = vy + 2.0f * (w * uvy + uuvy);
    float rz = vz + 2.0f * (w * uvz + uuvz);
    float a2 = rx * rx + ry * ry + rz * rz;

    eqiv[n * ROWS_PER_Q + k]         = make_float4(-2.0f * rx, -2.0f * ry, -2.0f * rz, a2);
    eqiv[n * ROWS_PER_Q + K_SYM + k] = make_float4( 2.0f * rx,  2.0f * ry,  2.0f * rz, a2);
}

// ---------------------------------------------------------------------------
// Kernel 2: pack FZ points as (bx, by, bz, b2); init per-query best keys.
// ---------------------------------------------------------------------------
__global__ void prep_pts(const float* __restrict__ s2,
                         float4* __restrict__ s2b,
                         unsigned long long* __restrict__ qbest) {
    int m = blockIdx.x * blockDim.x + threadIdx.x;
    if (m < M_FZ) {
        float bx = s2[m * 3 + 0], by = s2[m * 3 + 1], bz = s2[m * 3 + 2];
        s2b[m] = make_float4(bx, by, bz, bx * bx + by * by + bz * bz);
    }
    if (m < N_QUERY) qbest[m] = ~0ull;
}

// ---------------------------------------------------------------------------
// Kernel 3: WMMA argmin. One wave32 per 16-row tile of the 98304-row eqiv
// matrix; loops 256 16-col tiles of the 4096 FZ points held in LDS.
// v_wmma_f32_16x16x4_f32: A = 16x4 rows (-2x,-2y,-2z,0), B = 4x16 cols.
// Hot loop keeps only a per-slot f32 running min (cmp + dual cndmask);
// sortable-key packing and 64-bit mins are deferred to the epilogue.
// ---------------------------------------------------------------------------
__global__ void __launch_bounds__(256)
argmin_kernel(const float4* __restrict__ eqiv,
              const float4* __restrict__ s2b,
              unsigned long long* __restrict__ qbest) {
    extern __shared__ float4 sh[];          // M_FZ * 16B = 64 KB (of 320 KB WGP LDS)
    for (int i = threadIdx.x; i < M_FZ; i += blockDim.x) sh[i] = s2b[i];
    __syncthreads();

    const int lane = threadIdx.x & 31;
    const int wid  = threadIdx.x >> 5;
    const int tile = blockIdx.x * WAVES_PER_BLOCK + wid;
    const int rowbase = tile * 16;
    const int nq    = rowbase / ROWS_PER_Q;       // 48 = 3*16: tiles never straddle queries
    const int kbase = rowbase % ROWS_PER_Q;
    const int col   = lane & 15;
    const int hi    = (lane >> 4) & 1;

    // A operand (16x4 f32, 2 VGPRs): low lanes hold K={0,1}, high lanes K={2,3}
    float4 ar = eqiv[rowbase + col];
    v2f a;
    a.x = hi ? ar.z : ar.x;    // K=0 -> -2x ; K=2 -> -2z
    a.y = hi ? 0.0f : ar.y;    // K=1 -> -2y ; K=3 -> 0 (pad)

    // a2 for the 8 rows this lane's accumulator slice covers (M = v + 8*hi)
    float a2v[8];
#pragma unroll
    for (int v = 0; v < 8; ++v) a2v[v] = eqiv[rowbase + v + hi * 8].w;

    // Per-slot running min: value + m-index. Strict < keeps the first
    // (smallest m) occurrence, matching jnp.argmin tie-breaking in-slot.
    float best_s[8];
    unsigned int best_m[8];
#pragma unroll
    for (int v = 0; v < 8; ++v) {
        best_s[v] = __int_as_float(0x7F800000);   // +inf
        best_m[v] = 0u;
    }

#pragma unroll 2
    for (int mt = 0; mt < MTILES; ++mt) {
        float4 bp = sh[mt * 16 + col];
        v2f b;
        b.x = hi ? bp.z : bp.x;   // K=0 -> bx ; K=2 -> bz
        b.y = hi ? 0.0f : bp.y;   // K=1 -> by ; K=3 -> pad (multiplied by 0)

        v8f acc = {0.f, 0.f, 0.f, 0.f, 0.f, 0.f, 0.f, 0.f};
        acc = __builtin_amdgcn_wmma_f32_16x16x4_f32(
            /*neg_a=*/false, a, /*neg_b=*/false, b,
            /*c_mod=*/(short)0, acc, /*reuse_a=*/false, /*reuse_b=*/false);

        const float b2 = bp.w;
        const unsigned int m = (unsigned int)(mt * 16 + col);
#pragma unroll
        for (int v = 0; v < 8; ++v) {
            float d2 = (a2v[v] + b2) + acc[v];    // matches a2 + b2 - 2*dots
            bool lt = d2 < best_s[v];
            best_s[v] = lt ? d2 : best_s[v];
            best_m[v] = lt ? m : best_m[v];
        }
    }

    // Epilogue: pack sortable (value, flat-idx) keys once; reduce.
    unsigned long long best = ~0ull;
#pragma unroll
    for (int v = 0; v < 8; ++v) {
        unsigned int kflat = (unsigned int)(kbase + v + hi * 8) * (unsigned int)M_FZ;
        unsigned int ub = __float_as_uint(best_s[v]);
        ub ^= (ub & 0x80000000u) ? 0xFFFFFFFFu : 0x80000000u;   // sortable float
        unsigned long long key =
            ((unsigned long long)ub << 32) | (unsigned long long)(kflat + best_m[v]);
        best = best < key ? best : key;   // ties -> smallest flat idx (jnp.argmin)
    }

    // wave32 reduction
#pragma unroll
    for (int off = 16; off > 0; off >>= 1) {
        unsigned long long o = __shfl_xor(best, off, 32);
        best = best < o ? best : o;
    }
    if (lane == 0) atomicMin(qbest + nq, best);
}

// ---------------------------------------------------------------------------
// Kernel 4: ids[n] = argmin_flat % M
// ---------------------------------------------------------------------------
__global__ void extract_ids(const unsigned long long* __restrict__ qbest,
                            int* __restrict__ ids) {
    int n = blockIdx.x * blockDim.x + threadIdx.x;
    if (n < N_QUERY)
        ids[n] = (int)((unsigned int)(qbest[n] & 0xFFFFFFFFull) % (unsigned int)M_FZ);
}

// ---------------------------------------------------------------------------
// Kernel 5: out = covmat[ids][:, ids]   (row id is a uniform scalar load;
// output is pure streaming -> nontemporal store keeps L2 for covmat rows)
// ---------------------------------------------------------------------------
__global__ void gather_kernel(const float* __restrict__ cov,
                              const int* __restrict__ ids,
                              float* __restrict__ out) {
    int row  = blockIdx.x >> 3;                       // 2048 rows
    int colc = ((blockIdx.x & 7) << 8) + threadIdx.x; // 8 blocks * 256 = 2048 cols
    int r = ids[row];
    int c = ids[colc];
    float val = cov[(size_t)r * M_FZ + c];
    __builtin_nontemporal_store(val, out + (size_t)row * N_QUERY + colc);
}

// ---------------------------------------------------------------------------
extern "C" void kernel_launch(void* const* d_in, const int* in_sizes, int n_in,
                              void* d_out, int out_size, void* d_ws, size_t ws_size,
                              hipStream_t stream) {
    const float* qpts  = (const float*)d_in[0];   // [2048,3]
    const float* s2    = (const float*)d_in[1];   // [4096,3]
    const float* quats = (const float*)d_in[2];   // [24,4]
    const float* cov   = (const float*)d_in[3];   // [4096,4096]
    float* out = (float*)d_out;

    char* ws = (char*)d_ws;
    float4* eqiv = (float4*)(ws);                                   // 98304*16 = 1572864 B
    float4* s2b  = (float4*)(ws + 1572864);                         // 65536 B
    unsigned long long* qbest = (unsigned long long*)(ws + 1638400);// 16384 B
    int* ids = (int*)(ws + 1654784);                                // 8192 B

    prep_rows<<<(N_QUERY * K_SYM + 255) / 256, 256, 0, stream>>>(qpts, quats, eqiv);
    prep_pts<<<(M_FZ + 255) / 256, 256, 0, stream>>>(s2, s2b, qbest);
    argmin_kernel<<<NTILES / WAVES_PER_BLOCK, 256, M_FZ * sizeof(float4), stream>>>(
        eqiv, s2b, qbest);
    extract_ids<<<(N_QUERY + 255) / 256, 256, 0, stream>>>(qbest, ids);
    gather_kernel<<<N_QUERY * (N_QUERY / 256), 256, 0, stream>>>(cov, ids, out);
}